// HOSHeadTemplate_6597069767493
// MI455X (gfx1250) — compile-verified
//
#include <hip/hip_runtime.h>

// Problem constants from the reference: B=8, H=W=512, C=3, CODE_SIZE=8, QUAD=4
#define HWP   262144   // H*W
#define NBAT  8
#define NCH   3

typedef float v2f __attribute__((ext_vector_type(2)));
typedef float v8f __attribute__((ext_vector_type(8)));

// Exact wave32 sum using the CDNA5 matrix pipe.
// A (16x4 f32): lane l, vgpr v -> A[l%16][2*(l/16)+v].  Put x in vgpr0, 0 in vgpr1:
//   rowsum_m = x_m + x_{m+16}.  With B = ones (layout-independent), D[m][n] = rowsum_m.
// D layout: lanes 0-15 vgpr d = D[d][n], lanes 16-31 vgpr d = D[d+8][n], so
//   sum(d[0..7]) gives Sum_{m=0..7} on lanes 0-15 and Sum_{m=8..15} on lanes 16-31;
//   one xor-16 shuffle completes the 32-lane total (EXEC must be all 1s at call site).
__device__ __forceinline__ float wave_sum32(float x) {
  v2f a; a[0] = x;    a[1] = 0.0f;
  v2f b; b[0] = 1.0f; b[1] = 1.0f;
  v8f c = {};
  v8f d = __builtin_amdgcn_wmma_f32_16x16x4_f32(false, a, false, b, (short)0, c,
                                                false, false);
  float s = d[0] + d[1] + d[2] + d[3] + d[4] + d[5] + d[6] + d[7];
  s += __shfl_xor(s, 16, 32);
  return s;
}

// One block = 2048 consecutive pixels of one image (128 blocks/image * 8 images = 1024 blocks).
// Each thread handles 2 groups of 4 consecutive pixels -> all loads are float4 / coalesced.
__global__ __launch_bounds__(256) void hos_loss_partials(
    const float* __restrict__ cls,  // (B,H,W,3) channel-last
    const float* __restrict__ box,  // (B,N,8)
    const float* __restrict__ spa,  // (B,N,4)
    const float* __restrict__ hm,   // (B,3,H,W)
    const float* __restrict__ lbl,  // (B,N,8)
    const float* __restrict__ ql,   // (B,N,4)
    float* __restrict__ ws) {
  const int b      = blockIdx.x >> 7;          // image index
  const int chunk  = (blockIdx.x & 127) << 11; // *2048 pixels within image
  const float*  hmb     = hm + (size_t)b * (NCH * HWP);
  const size_t  pixBase = (size_t)b * HWP;

  // acc: [0..2]=cls S per channel, [3..5]=cls N per channel,
  //      [6]=smoothL1 masked sum, [7]=quadrant BCE masked sum, [8]=n_pos
  float acc[9];
#pragma unroll
  for (int i = 0; i < 9; ++i) acc[i] = 0.0f;

#pragma unroll
  for (int k = 0; k < 2; ++k) {
    const int n = chunk + (k << 10) + (threadIdx.x << 2);  // 4 consecutive pixels

    // heatmap targets: 3 coalesced float4 plane loads
    float ta[12];
    *(float4*)&ta[0] = *(const float4*)(hmb + 0 * HWP + n);
    *(float4*)&ta[4] = *(const float4*)(hmb + 1 * HWP + n);
    *(float4*)&ta[8] = *(const float4*)(hmb + 2 * HWP + n);

    // logits: 12 consecutive floats (16B aligned since n%4==0)
    float la[12];
    const float4* cp = (const float4*)(cls + (pixBase + (size_t)n) * 3);
    *(float4*)&la[0] = cp[0];
    *(float4*)&la[4] = cp[1];
    *(float4*)&la[8] = cp[2];

#pragma unroll
    for (int j = 0; j < 4; ++j) {
      bool posj = false;
#pragma unroll
      for (int c = 0; c < 3; ++c) {
        const float t = ta[c * 4 + j];     // target in [0,1]
        const float l = la[j * 3 + c];     // logit
        const float w = (t > 0.0f || t == 0.0f) ? 1.0f : 0.0f;  // pos+neg
        posj = posj || (t > 0.0f);
        const float p   = 1.0f / (1.0f + __expf(-l));           // sigmoid
        const float aw  = 0.75f - 0.5f * t;                     // t*A + (1-t)*(1-A), A=0.25
        const float pt  = t + p - 2.0f * t * p;                 // t*(1-p)+(1-t)*p
        const float bce = fmaxf(l, 0.0f) - l * t + __logf(1.0f + __expf(-fabsf(l)));
        acc[c]     += aw * pt * pt * bce * w;
        acc[3 + c] += w;
      }

      if (posj) {  // ~6.7% of pixels: predicate the 96B of regression traffic
        acc[8] += 1.0f;
        const size_t pix = pixBase + (size_t)(n + j);

        // smooth L1 over 8 components
        const float4* bp = (const float4*)box + pix * 2;
        const float4* lp = (const float4*)lbl + pix * 2;
        const float4 b0 = bp[0], b1 = bp[1], g0 = lp[0], g1 = lp[1];
        const float db[8] = {b0.x - g0.x, b0.y - g0.y, b0.z - g0.z, b0.w - g0.w,
                             b1.x - g1.x, b1.y - g1.y, b1.z - g1.z, b1.w - g1.w};
        float s = 0.0f;
#pragma unroll
        for (int e = 0; e < 8; ++e) {
          const float ad = fabsf(db[e]);
          s += (ad < 1.0f) ? 0.5f * db[e] * db[e] : (ad - 0.5f);
        }
        acc[6] += s;

        // quadrant BCE over 4 components
        const float4 q = ((const float4*)spa)[pix];
        const float4 y = ((const float4*)ql)[pix];
        const float qa[4] = {q.x, q.y, q.z, q.w};
        const float ya[4] = {y.x, y.y, y.z, y.w};
        float sb = 0.0f;
#pragma unroll
        for (int e = 0; e < 4; ++e) {
          const float pq   = fminf(fmaxf(ya[e] * qa[e], 0.0f), 1.0f);
          const float lpv  = fmaxf(__logf(fmaxf(pq, 1e-12f)), -100.0f);
          const float l1pv = fmaxf(__logf(1.0f - fminf(pq, 1.0f - 1e-12f)), -100.0f);
          sb -= ya[e] * lpv + (1.0f - ya[e]) * l1pv;
        }
        acc[7] += sb;
      }
    }
  }

  // --- block reduction: WMMA wave sums, then 8-wave combine in LDS ---
  __shared__ float red[8][9];
  const int wv = threadIdx.x >> 5;
#pragma unroll
  for (int i = 0; i < 9; ++i) acc[i] = wave_sum32(acc[i]);  // full EXEC here
  if ((threadIdx.x & 31) == 0) {
#pragma unroll
    for (int i = 0; i < 9; ++i) red[wv][i] = acc[i];
  }
  __syncthreads();
  if (threadIdx.x < 9) {
    float v = 0.0f;
#pragma unroll
    for (int w2 = 0; w2 < 8; ++w2) v += red[w2][threadIdx.x];
    const int t = threadIdx.x;
    // ws layout: [0..23]=S[c*8+b], [24..47]=N[c*8+b], [48]=sl1, [49]=bce_q, [50]=n_pos
    const int bin = (t < 3) ? (t * NBAT + b)
                  : (t < 6) ? (24 + (t - 3) * NBAT + b)
                            : (48 + (t - 6));
    unsafeAtomicAdd(&ws[bin], v);
  }
}

// Single-wave finalize: per-(c,b) normalized cls sum via one more WMMA reduction.
__global__ __launch_bounds__(32) void hos_loss_finalize(const float* __restrict__ ws,
                                                        float* __restrict__ out) {
  const int t = threadIdx.x;
  const float v = (t < 24) ? ws[t] / fmaxf(ws[24 + t], 1.0f) : 0.0f;
  const float clssum = wave_sum32(v);
  if (t == 0) {
    const float np = fmaxf(ws[50], 1.0f);
    // cls/B + 8*LOC_WEIGHT*sl1/(8*npos) + bce/npos  ==  cls*0.125 + 0.25*sl1/np + bce/np
    out[0] = clssum * 0.125f + 0.25f * ws[48] / np + ws[49] / np;
  }
}

extern "C" void kernel_launch(void* const* d_in, const int* in_sizes, int n_in,
                              void* d_out, int out_size, void* d_ws, size_t ws_size,
                              hipStream_t stream) {
  const float* cls = (const float*)d_in[0];  // cls_preds  (8,512,512,3)
  const float* box = (const float*)d_in[1];  // box_preds  (8,N,8)
  const float* spa = (const float*)d_in[2];  // spa_preds  (8,N,4)
  const float* hm  = (const float*)d_in[3];  // heatmaps   (8,3,512,512)
  const float* lbl = (const float*)d_in[4];  // hos_box_labels
  const float* ql  = (const float*)d_in[5];  // quadrant_labels
  float* ws = (float*)d_ws;

  hipMemsetAsync(ws, 0, 64 * sizeof(float), stream);  // zero accumulators every call
  hos_loss_partials<<<1024, 256, 0, stream>>>(cls, box, spa, hm, lbl, ql, ws);
  hos_loss_finalize<<<1, 32, 0, stream>>>(ws, (float*)d_out);
}